// Attention_65369402245756
// MI455X (gfx1250) — compile-verified
//
#include <hip/hip_runtime.h>
#include <math.h>

#define B_ 32
#define S_ 4096
#define D_ 256
#define CH 512   // S-chunk per context block

typedef float v2f __attribute__((ext_vector_type(2)));
typedef float v8f __attribute__((ext_vector_type(8)));

// ---------------------------------------------------------------------------
// Kernel 1: energy[b,s] = dot(key[b,s,:], token[b,:])
// Each wave computes 16 rows via V_WMMA_F32_16X16X4_F32, K-loop of 64 steps.
// B fragment = token broadcast across all 16 N columns, so every column of the
// 16x16 accumulator holds the same 16 energies.
// Block: 256 threads = 8 waves -> 128 rows per block. Grid: (B, S/128).
// ---------------------------------------------------------------------------
__global__ __launch_bounds__(256) void energy_kernel(
    const float* __restrict__ key, const float* __restrict__ token,
    float* __restrict__ energy) {
  const int b = blockIdx.x;
  __shared__ float tok[D_];
  tok[threadIdx.x] = token[b * D_ + threadIdx.x];
  __syncthreads();

  const int lane = threadIdx.x & 31;
  const int wave = threadIdx.x >> 5;
  const int r0 = blockIdx.y * 128 + wave * 16;
  const int row = r0 + (lane & 15);
  const int kShift = (lane >> 4) * 2;  // lanes 0-15: K=0,1 ; lanes 16-31: K=2,3
  const float* krow = key + ((size_t)b * S_ + row) * D_;

  v8f acc = {};
#pragma unroll 8
  for (int k0 = 0; k0 < D_; k0 += 4) {
    const float2 ka = *(const float2*)(krow + k0 + kShift);
    const float2 tb = *(const float2*)(&tok[k0 + kShift]);
    v2f a = {ka.x, ka.y};
    v2f bb = {tb.x, tb.y};
    acc = __builtin_amdgcn_wmma_f32_16x16x4_f32(
        /*neg_a=*/false, a, /*neg_b=*/false, bb,
        /*c_mod=*/(short)0, acc, /*reuse_a=*/false, /*reuse_b=*/false);
  }

  // C/D layout: VGPR r holds row M=r (lanes 0-15) and M=r+8 (lanes 16-31);
  // all columns identical (B broadcast) -> lanes 0 and 16 export 16 energies.
  float* eout = energy + (size_t)b * S_ + r0;
  if (lane == 0) {
#pragma unroll
    for (int r = 0; r < 8; ++r) eout[r] = acc[r];
  } else if (lane == 16) {
#pragma unroll
    for (int r = 0; r < 8; ++r) eout[8 + r] = acc[r];
  }
}

// ---------------------------------------------------------------------------
// Kernel 2: per-batch softmax scalars (max, sum-of-exp over ALL S positions:
// the reference masks AFTER softmax). Also zeroes the output row so kernel 3
// can accumulate with atomics. Grid: B blocks of 256 threads.
// ---------------------------------------------------------------------------
__global__ __launch_bounds__(256) void softmax_scalars_kernel(
    const float* __restrict__ energy, float* __restrict__ scal,
    float* __restrict__ out) {
  const int b = blockIdx.x;
  const int t = threadIdx.x;
  __shared__ float red[256];
  const float* e = energy + (size_t)b * S_;

  float loc[S_ / 256];
  float m = -INFINITY;
#pragma unroll
  for (int i = 0; i < S_ / 256; ++i) {
    loc[i] = e[t + 256 * i];
    m = fmaxf(m, loc[i]);
  }
  red[t] = m;
  __syncthreads();
  for (int s = 128; s > 0; s >>= 1) {
    if (t < s) red[t] = fmaxf(red[t], red[t + s]);
    __syncthreads();
  }
  m = red[0];
  __syncthreads();

  float sum = 0.f;
#pragma unroll
  for (int i = 0; i < S_ / 256; ++i) sum += __expf(loc[i] - m);
  red[t] = sum;
  __syncthreads();
  for (int s = 128; s > 0; s >>= 1) {
    if (t < s) red[t] += red[t + s];
    __syncthreads();
  }
  if (t == 0) {
    scal[2 * b] = m;
    scal[2 * b + 1] = red[0];
  }
  out[b * D_ + t] = 0.f;  // init for atomic accumulation in kernel 3
}

// ---------------------------------------------------------------------------
// Kernel 3: context[b,d] = sum_s w[b,s] * value[b,s,d], with
// w = (s < len) ? exp(e-m)/Z : 1e-9. Weights staged in LDS, value streamed
// with b128 loads + global_prefetch, partials folded in with f32 atomics.
// Grid: (B, S/CH) blocks of 256 threads; thread owns 4 contiguous columns.
// ---------------------------------------------------------------------------
__global__ __launch_bounds__(256) void context_kernel(
    const float* __restrict__ value, const float* __restrict__ energy,
    const float* __restrict__ scal, const int* __restrict__ lens,
    float* __restrict__ out) {
  const int b = blockIdx.x;
  const int s0 = blockIdx.y * CH;
  const int t = threadIdx.x;

  __shared__ float w[CH];
  const float m = scal[2 * b];
  const float invZ = 1.0f / scal[2 * b + 1];
  const int len = lens[b];
  const float* e = energy + (size_t)b * S_ + s0;
#pragma unroll
  for (int j = t; j < CH; j += 256) {
    const int s = s0 + j;
    w[j] = (s < len) ? __expf(e[j] - m) * invZ : 1e-9f;
  }
  __syncthreads();

  const int dcol = (t & 63) * 4;   // 64 column-groups of float4 cover D=256
  const int jstart = t >> 6;       // 4 row phases
  const float* vb = value + ((size_t)b * S_ + s0) * D_ + dcol;

  float4 acc = {0.f, 0.f, 0.f, 0.f};
  for (int j = jstart; j < CH; j += 4) {
    __builtin_prefetch(vb + (size_t)(j + 16) * D_, 0, 0);  // global_prefetch_b8
    const float4 v = *(const float4*)(vb + (size_t)j * D_);
    const float ww = w[j];
    acc.x = fmaf(ww, v.x, acc.x);
    acc.y = fmaf(ww, v.y, acc.y);
    acc.z = fmaf(ww, v.z, acc.z);
    acc.w = fmaf(ww, v.w, acc.w);
  }

  float* o = out + b * D_ + dcol;
  atomicAdd(o + 0, acc.x);
  atomicAdd(o + 1, acc.y);
  atomicAdd(o + 2, acc.z);
  atomicAdd(o + 3, acc.w);
}

// ---------------------------------------------------------------------------
extern "C" void kernel_launch(void* const* d_in, const int* in_sizes, int n_in,
                              void* d_out, int out_size, void* d_ws,
                              size_t ws_size, hipStream_t stream) {
  const float* key = (const float*)d_in[0];
  const float* value = (const float*)d_in[1];
  const float* token = (const float*)d_in[2];
  const int* lens = (const int*)d_in[3];
  float* out = (float*)d_out;

  // Workspace: energy [B*S] floats (512 KiB) + per-batch (max, Z) scalars.
  float* energy = (float*)d_ws;
  float* scal = energy + (size_t)B_ * S_;

  energy_kernel<<<dim3(B_, S_ / 128), 256, 0, stream>>>(key, token, energy);
  softmax_scalars_kernel<<<dim3(B_), 256, 0, stream>>>(energy, scal, out);
  context_kernel<<<dim3(B_, S_ / CH), 256, 0, stream>>>(value, energy, scal,
                                                        lens, out);
}